// WeightedRGCN_67319317398089
// MI455X (gfx1250) — compile-verified
//
#include <hip/hip_runtime.h>
#include <stddef.h>

typedef __attribute__((ext_vector_type(2))) float v2f;
typedef __attribute__((ext_vector_type(8))) float v8f;

static constexpr int FD = 128;  // feature dim == hidden dim

// ---------------------------------------------------------------------------
// zero fill (graph-capture-safe replacement for hipMemsetAsync)
// ---------------------------------------------------------------------------
__global__ void zero_f32_kernel(float* __restrict__ p, size_t n) {
  size_t i = (size_t)blockIdx.x * blockDim.x + threadIdx.x;
  size_t stride = (size_t)gridDim.x * blockDim.x;
  for (; i < n; i += stride) p[i] = 0.0f;
}

// ---------------------------------------------------------------------------
// fold the three user-side Wr matrices / biases into one (W_DIRECT/AUTHOR/SOCIAL)
// ---------------------------------------------------------------------------
__global__ void combine_wr_kernel(const float* __restrict__ Wr_d, const float* __restrict__ Wr_a,
                                  const float* __restrict__ Wr_s, const float* __restrict__ bl_d,
                                  const float* __restrict__ bl_a, const float* __restrict__ bl_s,
                                  float* __restrict__ Wru, float* __restrict__ bu) {
  int i = blockIdx.x * blockDim.x + threadIdx.x;
  if (i < FD * FD) Wru[i] = 1.75f * Wr_d[i] + 0.7f * Wr_a[i] + 0.3f * Wr_s[i];
  if (i < FD)      bu[i]  = 1.75f * bl_d[i] + 0.7f * bl_a[i] + 0.3f * bl_s[i];
}

// ---------------------------------------------------------------------------
// edge scatter: one wave32 per edge, float4 gather per lane (32*16B = 512B row),
// f32 atomics into agg[dst] (agg fits in the 192MB L2 -> atomics resolve on-chip)
// ---------------------------------------------------------------------------
__global__ void edge_scatter_kernel(const float* __restrict__ xsrc, const int* __restrict__ src,
                                    const int* __restrict__ dst, float* __restrict__ agg,
                                    float* __restrict__ cnt, int E) {
  int e = blockIdx.x * (blockDim.x >> 5) + (threadIdx.x >> 5);
  if (e >= E) return;
  int lane = threadIdx.x & 31;

  // warm L2 for a future edge's gather row (global_prefetch_b8)
  int e2 = e + 2048;
  if (e2 < E) {
    int s2 = src[e2];
    __builtin_prefetch(xsrc + (size_t)s2 * FD + lane * 4, 0, 1);
  }

  int s = src[e];
  int d = dst[e];
  const float4 v = ((const float4*)(xsrc + (size_t)s * FD))[lane];
  float* ag = agg + (size_t)d * FD + lane * 4;
  atomicAdd(ag + 0, v.x);
  atomicAdd(ag + 1, v.y);
  atomicAdd(ag + 2, v.z);
  atomicAdd(ag + 3, v.w);
  if (lane == 0) atomicAdd(cnt + d, 1.0f);
}

// ---------------------------------------------------------------------------
// Stage a 128x128 f32 weight matrix into LDS in K-pair-interleaved layout:
//   lds[p*256 + n*2 + j] = W[(2p + j)*128 + n]   (p = K/2 pair row)
// so each WMMA B fragment {B[k][n], B[k+1][n]} is a single aligned ds_load_b64.
// ---------------------------------------------------------------------------
__device__ __forceinline__ void stage_weights_paired(const float* __restrict__ W,
                                                     float* __restrict__ lds) {
  for (int pr = threadIdx.x; pr < (FD / 2) * FD; pr += blockDim.x) {
    int p = pr >> 7;   // pair row (0..63)
    int n = pr & 127;  // column
    v2f w;
    w.x = W[(2 * p) * FD + n];
    w.y = W[(2 * p + 1) * FD + n];
    *(v2f*)(lds + p * 2 * FD + n * 2) = w;
  }
}

// ---------------------------------------------------------------------------
// One K=64 pass (K 0..63, 16 k-steps) of a 16x128 output tile via V_WMMA_F32_16X16X4_F32.
// A layout (16x4 f32): lane m = lane&15, K = 4*t + 2*(lane>>4) + {0,1}
// B layout (4x16 f32): lane n = lane&15, K row = 4*t + 2*(lane>>4) + vgpr
// All 16 A fragments preloaded up-front (no per-kstep load-wait stall);
// sc folds (edge-type weight / mean count) into the A operand.
// ---------------------------------------------------------------------------
__device__ __forceinline__ void wmma_pass_128(const float* __restrict__ A, float sc,
                                              const float* __restrict__ ldsP, int row, int half,
                                              int lm, v8f acc[8]) {
  const float* ap = A + (size_t)row * FD + 2 * half;
  v2f areg[16];
#pragma unroll
  for (int t = 0; t < 16; ++t) areg[t] = *(const v2f*)(ap + 4 * t);

#pragma unroll
  for (int t = 0; t < 16; ++t) {
    v2f a;
    a.x = areg[t].x * sc;
    a.y = areg[t].y * sc;
    const int p = 2 * t + half;  // K-pair row
    const float* bp = ldsP + p * 2 * FD + lm * 2;
#pragma unroll
    for (int c = 0; c < 8; ++c) {
      v2f b = *(const v2f*)(bp + c * 32);  // one ds_load_b64, WMMA-ready pair
      acc[c] = __builtin_amdgcn_wmma_f32_16x16x4_f32(false, a, false, b, (short)0, acc[c],
                                                     false, false);
    }
  }
}

// out[tile16 x 128] (+)= (agg/cnt * wscale) @ Wl  [ + xdst @ Wr + bias, relu ]
template <bool LOAD_C, bool SECOND, bool RELU>
__global__ __launch_bounds__(256) void sage_gemm_kernel(
    const float* __restrict__ agg, const float* __restrict__ cnt, float wscale,
    const float* __restrict__ Wl, const float* __restrict__ xdst, const float* __restrict__ Wr,
    const float* __restrict__ bias, float* __restrict__ out, int n) {
  __shared__ float ldsB[FD * FD];  // 64KB weight stage (320KB LDS per WGP on CDNA5)

  stage_weights_paired(Wl, ldsB);
  __syncthreads();

  const int tile = blockIdx.x * (blockDim.x >> 5) + (threadIdx.x >> 5);
  const int lane = threadIdx.x & 31;
  const int half = lane >> 4;
  const int lm = lane & 15;
  const int rows = tile * 16;
  const bool valid = rows < n;  // wave-uniform -> EXEC all-1s inside (WMMA requirement)
  const int row = valid ? (rows + lm) : lm;

  v8f acc[8];
  if (valid) {
    if (LOAD_C) {
#pragma unroll
      for (int c = 0; c < 8; ++c)
#pragma unroll
        for (int v = 0; v < 8; ++v)
          acc[c][v] = out[(size_t)(rows + v + 8 * half) * FD + c * 16 + lm];
    } else {
#pragma unroll
      for (int c = 0; c < 8; ++c)
#pragma unroll
        for (int v = 0; v < 8; ++v) acc[c][v] = 0.0f;
    }
    const float sc = wscale / fmaxf(cnt[row], 1.0f);  // mean + edge-type weight folded into A
    wmma_pass_128(agg, sc, ldsB, row, half, lm, acc);
  }

  if (SECOND) {
    __syncthreads();
    stage_weights_paired(Wr, ldsB);
    __syncthreads();
    if (valid) wmma_pass_128(xdst, 1.0f, ldsB, row, half, lm, acc);
  }

  if (valid) {
#pragma unroll
    for (int c = 0; c < 8; ++c) {
      const float bz = SECOND ? bias[c * 16 + lm] : 0.0f;
#pragma unroll
      for (int v = 0; v < 8; ++v) {
        float val = acc[c][v] + bz;
        if (RELU) val = fmaxf(val, 0.0f);
        out[(size_t)(rows + v + 8 * half) * FD + c * 16 + lm] = val;
      }
    }
  }
}

// ---------------------------------------------------------------------------
extern "C" void kernel_launch(void* const* d_in, const int* in_sizes, int n_in, void* d_out,
                              int out_size, void* d_ws, size_t ws_size, hipStream_t stream) {
  const float* x_user = (const float*)d_in[0];
  const float* x_post = (const float*)d_in[1];
  const int* re_src = (const int*)d_in[2];
  const int* re_dst = (const int*)d_in[3];
  const int* fb_src = (const int*)d_in[4];
  const int* fb_dst = (const int*)d_in[5];
  const int* soc_src = (const int*)d_in[6];
  const int* soc_dst = (const int*)d_in[7];
  const int* eng_src = (const int*)d_in[8];
  const int* eng_dst = (const int*)d_in[9];
  const float* Wl_d = (const float*)d_in[10];
  const float* bl_d = (const float*)d_in[11];
  const float* Wr_d = (const float*)d_in[12];
  const float* Wl_a = (const float*)d_in[13];
  const float* bl_a = (const float*)d_in[14];
  const float* Wr_a = (const float*)d_in[15];
  const float* Wl_s = (const float*)d_in[16];
  const float* bl_s = (const float*)d_in[17];
  const float* Wr_s = (const float*)d_in[18];
  const float* Wl_p = (const float*)d_in[19];
  const float* bl_p = (const float*)d_in[20];
  const float* Wr_p = (const float*)d_in[21];

  const int E_re = in_sizes[2];
  const int E_fb = in_sizes[4];
  const int E_soc = in_sizes[6];
  const int E_eng = in_sizes[8];
  const int n_user = in_sizes[0] / FD;  // 100000
  const int n_post = in_sizes[1] / FD;  // 200000

  // workspace: one reusable agg buffer sized for the largest node set (~102MB,
  // fits in MI455X's 192MB L2 so scatter atomics stay on-chip) + counts + Wru/bu
  float* agg = (float*)d_ws;
  float* cnt = agg + (size_t)n_post * FD;
  float* Wru = cnt + n_post;
  float* bu = Wru + FD * FD;

  float* out_user = (float*)d_out;
  float* out_post = out_user + (size_t)n_user * FD;

  combine_wr_kernel<<<(FD * FD + 255) / 256, 256, 0, stream>>>(Wr_d, Wr_a, Wr_s, bl_d, bl_a,
                                                               bl_s, Wru, bu);

  const int ZB = 4096;  // grid-stride zero fill blocks
  auto gemm_blocks = [](int n) { return (n / 16 + 7) / 8; };
  auto scat_blocks = [](int E) { return (E + 7) / 8; };

  // --- phase 1: direct (posts -> users via reposts), weight 1.75 ------------
  zero_f32_kernel<<<ZB, 256, 0, stream>>>(agg, (size_t)n_user * FD);
  zero_f32_kernel<<<256, 256, 0, stream>>>(cnt, (size_t)n_user);
  edge_scatter_kernel<<<scat_blocks(E_re), 256, 0, stream>>>(x_post, re_src, re_dst, agg, cnt,
                                                             E_re);
  sage_gemm_kernel<false, false, false><<<gemm_blocks(n_user), 256, 0, stream>>>(
      agg, cnt, 1.75f, Wl_d, nullptr, nullptr, nullptr, out_user, n_user);

  // --- phase 2: author (posts -> users via feedback), weight 0.7 ------------
  zero_f32_kernel<<<ZB, 256, 0, stream>>>(agg, (size_t)n_user * FD);
  zero_f32_kernel<<<256, 256, 0, stream>>>(cnt, (size_t)n_user);
  edge_scatter_kernel<<<scat_blocks(E_fb), 256, 0, stream>>>(x_post, fb_src, fb_dst, agg, cnt,
                                                             E_fb);
  sage_gemm_kernel<true, false, false><<<gemm_blocks(n_user), 256, 0, stream>>>(
      agg, cnt, 0.7f, Wl_a, nullptr, nullptr, nullptr, out_user, n_user);

  // --- phase 3: social (users -> users), weight 0.3 + x_user@Wru + bu, relu -
  zero_f32_kernel<<<ZB, 256, 0, stream>>>(agg, (size_t)n_user * FD);
  zero_f32_kernel<<<256, 256, 0, stream>>>(cnt, (size_t)n_user);
  edge_scatter_kernel<<<scat_blocks(E_soc), 256, 0, stream>>>(x_user, soc_src, soc_dst, agg, cnt,
                                                              E_soc);
  sage_gemm_kernel<true, true, true><<<gemm_blocks(n_user), 256, 0, stream>>>(
      agg, cnt, 0.3f, Wl_s, x_user, Wru, bu, out_user, n_user);

  // --- phase 4: posts (users -> posts via engagement) + x_post@Wr_p, relu ---
  zero_f32_kernel<<<ZB, 256, 0, stream>>>(agg, (size_t)n_post * FD);
  zero_f32_kernel<<<256, 256, 0, stream>>>(cnt, (size_t)n_post);
  edge_scatter_kernel<<<scat_blocks(E_eng), 256, 0, stream>>>(x_user, eng_src, eng_dst, agg, cnt,
                                                              E_eng);
  sage_gemm_kernel<false, true, true><<<gemm_blocks(n_post), 256, 0, stream>>>(
      agg, cnt, 1.0f, Wl_p, x_post, Wr_p, bl_p, out_post, n_post);
}